// RingAttentionBottleneck_3547642987426
// MI455X (gfx1250) — compile-verified
//
#include <hip/hip_runtime.h>
#include <hip/hip_bf16.h>
#include <cstddef>

// ---------------------------------------------------------------------------
// RingAttentionBottleneck for MI455X (gfx1250, wave32)
//   B=4, C=256, H=W=64, NH=4, HD=64, RINGS={4,8,16}, NS=8, KOFF=24
//
// Pipeline:
//   1) qkv_wmma:  q/k/v = W{q,k,v} @ x + b. One wave computes 16x32 tiles of
//      ALL THREE outputs sharing the X (B-matrix) fragment: 6 WMMAs per K-step.
//      Weight tiles staged in LDS (padded stride 260 -> conflict-free b64 reads).
//   2) ring_attn: bilinear ring gather + per-pixel softmax (VALU/VMEM gather).
//   3) proj_wmma: y = x + Wo @ attn_out + bo. 16x64 tile/wave, 4 WMMAs/K-step,
//      Wo tile staged in LDS the same way.
// All f32 WMMA (V_WMMA_F32_16X16X4_F32): full reference precision on the matrix
// pipe. Working set (~84 MB) fits in 192 MB L2 -> GEMM FLOPs dominate.
// ---------------------------------------------------------------------------

typedef float v2f __attribute__((ext_vector_type(2)));
typedef float v8f __attribute__((ext_vector_type(8)));

#define CC   256
#define HH   64
#define WW   64
#define HWD  (HH * WW)          // 4096
#define BB   4
#define NHD  4
#define HDIM 64
#define KOFF 24
#define LDSW 260                // padded LDS row stride (floats): bank = 4*row

__device__ __constant__ float c_cos8[8] =
    {1.0f, 0.70710678f, 0.0f, -0.70710678f, -1.0f, -0.70710678f, 0.0f, 0.70710678f};
__device__ __constant__ float c_sin8[8] =
    {0.0f, 0.70710678f, 1.0f, 0.70710678f, 0.0f, -0.70710678f, -1.0f, -0.70710678f};
__device__ __constant__ float c_ring[3] = {4.0f, 8.0f, 16.0f};

__device__ __forceinline__ v8f wmma_f32(v2f a, v2f b, v8f c) {
    return __builtin_amdgcn_wmma_f32_16x16x4_f32(
        /*neg_a=*/false, a, /*neg_b=*/false, b,
        /*c_mod=*/(short)0, c, /*reuse_a=*/false, /*reuse_b=*/false);
}

// ---------------------------------------------------------------------------
// Kernel 1: fused q/k/v projection. blockDim = 128 (4 waves).
// Wave tile: M=16 out-channels x N=32 pixels, for q AND k AND v (B shared).
// A 16x4 f32 layout (ISA 7.12.2): lane<16 holds K={0,1}, lane>=16 K={2,3}.
// D 16x16 f32 layout: VGPR i -> M = i + 8*(lane>>4), N = lane&15.
// ---------------------------------------------------------------------------
__global__ void __launch_bounds__(128)
qkv_wmma(const float* __restrict__ x,
         const float* __restrict__ wq, const float* __restrict__ bq,
         const float* __restrict__ wk, const float* __restrict__ bk,
         const float* __restrict__ wv, const float* __restrict__ bv,
         float* __restrict__ q, float* __restrict__ k, float* __restrict__ v)
{
    __shared__ float sW[3][16][LDSW];                  // 48.75 KB of 320 KB/WGP

    const int lane = threadIdx.x & 31;
    const int wid  = threadIdx.x >> 5;                 // wave in block (0..3)
    const int n0   = (blockIdx.x * 4 + wid) * 32;      // 32-wide spatial tile
    const int m0   = blockIdx.y * 16;                  // out-channel tile
    const int b    = blockIdx.z;

    // Cooperative, coalesced fill of the three 16x256 weight tiles.
    {
        const float* Ws[3] = {wq, wk, wv};
        #pragma unroll
        for (int mtx = 0; mtx < 3; ++mtx) {
            const float* Wg = Ws[mtx] + (size_t)m0 * CC;
            for (int idx = threadIdx.x; idx < 16 * CC; idx += 128) {
                const int row = idx >> 8;              // 0..15
                const int col = idx & (CC - 1);        // 0..255 (consecutive per t)
                sW[mtx][row][col] = Wg[row * CC + col];
            }
        }
    }
    __syncthreads();

    const int mrow  = lane & 15;
    const int khalf = (lane >> 4) * 2;

    // B fragment base: x[b][kk + khalf + {0,1}][n0 + mrow (+16)]
    const float* Xc = x + ((size_t)b * CC + khalf) * HWD + (n0 + mrow);
    const float* Aq = &sW[0][mrow][khalf];
    const float* Ak = &sW[1][mrow][khalf];
    const float* Av = &sW[2][mrow][khalf];

    v8f aq0 = {}, aq1 = {}, ak0 = {}, ak1 = {}, av0 = {}, av1 = {};

    #pragma unroll 2
    for (int kk = 0; kk < CC; kk += 4) {
        // keep the streamed B operand ahead in cache (global_prefetch_b8)
        __builtin_prefetch(Xc + (size_t)(kk + 16) * HWD, 0, 1);

        v2f b0, b1, aq, ak, av;
        b0.x = Xc[(size_t)(kk + 0) * HWD];
        b0.y = Xc[(size_t)(kk + 1) * HWD];
        b1.x = Xc[(size_t)(kk + 0) * HWD + 16];
        b1.y = Xc[(size_t)(kk + 1) * HWD + 16];
        aq.x = Aq[kk]; aq.y = Aq[kk + 1];              // ds_load_b64, conflict-free
        ak.x = Ak[kk]; ak.y = Ak[kk + 1];
        av.x = Av[kk]; av.y = Av[kk + 1];

        aq0 = wmma_f32(aq, b0, aq0);  aq1 = wmma_f32(aq, b1, aq1);
        ak0 = wmma_f32(ak, b0, ak0);  ak1 = wmma_f32(ak, b1, ak1);
        av0 = wmma_f32(av, b0, av0);  av1 = wmma_f32(av, b1, av1);
    }

    const size_t obase = ((size_t)b * CC) * HWD;
    const int ncol  = n0 + (lane & 15);
    const int mbase = m0 + 8 * (lane >> 4);
    #pragma unroll
    for (int i = 0; i < 8; ++i) {
        const size_t row = (size_t)(mbase + i) * HWD;
        const float  biq = bq[mbase + i], bik = bk[mbase + i], biv = bv[mbase + i];
        q[obase + row + ncol]      = aq0[i] + biq;
        q[obase + row + ncol + 16] = aq1[i] + biq;
        k[obase + row + ncol]      = ak0[i] + bik;
        k[obase + row + ncol + 16] = ak1[i] + bik;
        v[obase + row + ncol]      = av0[i] + biv;
        v[obase + row + ncol + 16] = av1[i] + biv;
    }
}

// ---------------------------------------------------------------------------
// Kernel 2: ring-sampled attention. One wave per (b, head, pixel).
// Each lane owns 2 of the 64 head dims; q.k dots reduced via shfl_xor.
// V samples kept in registers so the gather pass runs once.
// ---------------------------------------------------------------------------
__global__ void __launch_bounds__(256)
ring_attn(const float* __restrict__ qf,
          const float* __restrict__ kf,
          const float* __restrict__ vf,
          float* __restrict__ out)
{
    const int lane = threadIdx.x & 31;
    const int gw   = blockIdx.x * 8 + (threadIdx.x >> 5);   // 0 .. 65535
    const int w    = gw & 63;
    const int h    = (gw >> 6) & 63;
    const int n    = (gw >> 12) & 3;
    const int b    = gw >> 14;

    const int    d0       = lane * 2;
    const size_t chanBase = ((size_t)b * CC + n * HDIM) * HWD;
    const int    hw       = h * WW + w;

    const float q0 = qf[chanBase + (size_t)(d0 + 0) * HWD + hw];
    const float q1 = qf[chanBase + (size_t)(d0 + 1) * HWD + hw];

    const float gx = -1.0f + 2.0f * (float)w / 63.0f;
    const float gy = -1.0f + 2.0f * (float)h / 63.0f;

    float scores[KOFF];
    float vs0[KOFF], vs1[KOFF];

    const float* k0p = kf + chanBase + (size_t)(d0 + 0) * HWD;
    const float* k1p = kf + chanBase + (size_t)(d0 + 1) * HWD;
    const float* v0p = vf + chanBase + (size_t)(d0 + 0) * HWD;
    const float* v1p = vf + chanBase + (size_t)(d0 + 1) * HWD;

    for (int ko = 0; ko < KOFF; ++ko) {
        const float r  = c_ring[ko >> 3];
        const float ox = r * c_cos8[ko & 7];
        const float oy = r * c_sin8[ko & 7];
        // reference: sx = clip(gx + off/(W/2), -1, 1); px = (sx+1)*0.5*(W-1)
        const float sx = fminf(fmaxf(gx + ox * (2.0f / (float)WW), -1.0f), 1.0f);
        const float sy = fminf(fmaxf(gy + oy * (2.0f / (float)HH), -1.0f), 1.0f);
        const float px = (sx + 1.0f) * 0.5f * (float)(WW - 1);
        const float py = (sy + 1.0f) * 0.5f * (float)(HH - 1);
        const float fx = floorf(px), fy = floorf(py);
        const int   x0 = (int)fx,    y0 = (int)fy;
        const float wx = px - fx,    wy = py - fy;
        const int x0c = min(max(x0, 0), WW - 1);
        const int x1c = min(max(x0 + 1, 0), WW - 1);
        const int y0c = min(max(y0, 0), HH - 1);
        const int y1c = min(max(y0 + 1, 0), HH - 1);
        const int i00 = y0c * WW + x0c, i01 = y0c * WW + x1c;
        const int i10 = y1c * WW + x0c, i11 = y1c * WW + x1c;
        const float w00 = (1.0f - wx) * (1.0f - wy);
        const float w01 = wx * (1.0f - wy);
        const float w10 = (1.0f - wx) * wy;
        const float w11 = wx * wy;

        const float k0 = w00 * k0p[i00] + w01 * k0p[i01] + w10 * k0p[i10] + w11 * k0p[i11];
        const float k1 = w00 * k1p[i00] + w01 * k1p[i01] + w10 * k1p[i10] + w11 * k1p[i11];
        vs0[ko]        = w00 * v0p[i00] + w01 * v0p[i01] + w10 * v0p[i10] + w11 * v0p[i11];
        vs1[ko]        = w00 * v1p[i00] + w01 * v1p[i01] + w10 * v1p[i10] + w11 * v1p[i11];

        float part = q0 * k0 + q1 * k1;              // partial dot over this lane's dims
        #pragma unroll
        for (int o = 16; o > 0; o >>= 1)             // wave32 butterfly reduction
            part += __shfl_xor(part, o, 32);
        scores[ko] = part * 0.125f;                  // SCALE = 64^-0.5
    }

    // softmax over the 24 offsets (wave-uniform, replicated per lane)
    float m = scores[0];
    #pragma unroll
    for (int ko = 1; ko < KOFF; ++ko) m = fmaxf(m, scores[ko]);
    float s = 0.0f;
    #pragma unroll
    for (int ko = 0; ko < KOFF; ++ko) { scores[ko] = __expf(scores[ko] - m); s += scores[ko]; }
    const float inv = 1.0f / s;

    float o0 = 0.0f, o1 = 0.0f;
    #pragma unroll
    for (int ko = 0; ko < KOFF; ++ko) {
        const float wgt = scores[ko] * inv;
        o0 += wgt * vs0[ko];
        o1 += wgt * vs1[ko];
    }
    out[chanBase + (size_t)(d0 + 0) * HWD + hw] = o0;
    out[chanBase + (size_t)(d0 + 1) * HWD + hw] = o1;
}

// ---------------------------------------------------------------------------
// Kernel 3: output projection + residual:  y = x + Wo @ attn_out + bo
// Wave tile: M=16 x N=64 (4 subtiles) -> A fragment reused 4x per K-step.
// ---------------------------------------------------------------------------
__global__ void __launch_bounds__(128)
proj_wmma(const float* __restrict__ ain,
          const float* __restrict__ x,
          const float* __restrict__ wo,
          const float* __restrict__ bo,
          float* __restrict__ y)
{
    __shared__ float sW[16][LDSW];                    // 16.6 KB

    const int lane = threadIdx.x & 31;
    const int wid  = threadIdx.x >> 5;
    const int n0   = (blockIdx.x * 4 + wid) * 64;     // 64-wide spatial tile
    const int m0   = blockIdx.y * 16;
    const int b    = blockIdx.z;

    {
        const float* Wg = wo + (size_t)m0 * CC;
        for (int idx = threadIdx.x; idx < 16 * CC; idx += 128) {
            const int row = idx >> 8;
            const int col = idx & (CC - 1);
            sW[row][col] = Wg[row * CC + col];
        }
    }
    __syncthreads();

    const int mrow  = lane & 15;
    const int khalf = (lane >> 4) * 2;

    const float* Ar = &sW[mrow][khalf];
    const float* Ac = ain + ((size_t)b * CC + khalf) * HWD + (n0 + mrow);

    v8f c0 = {}, c1 = {}, c2 = {}, c3 = {};

    #pragma unroll 2
    for (int kk = 0; kk < CC; kk += 4) {
        __builtin_prefetch(Ac + (size_t)(kk + 16) * HWD, 0, 1);

        v2f a, b0, b1, b2, b3;
        a.x  = Ar[kk]; a.y = Ar[kk + 1];              // ds_load_b64, conflict-free
        b0.x = Ac[(size_t)(kk + 0) * HWD +  0]; b0.y = Ac[(size_t)(kk + 1) * HWD +  0];
        b1.x = Ac[(size_t)(kk + 0) * HWD + 16]; b1.y = Ac[(size_t)(kk + 1) * HWD + 16];
        b2.x = Ac[(size_t)(kk + 0) * HWD + 32]; b2.y = Ac[(size_t)(kk + 1) * HWD + 32];
        b3.x = Ac[(size_t)(kk + 0) * HWD + 48]; b3.y = Ac[(size_t)(kk + 1) * HWD + 48];

        c0 = wmma_f32(a, b0, c0);
        c1 = wmma_f32(a, b1, c1);
        c2 = wmma_f32(a, b2, c2);
        c3 = wmma_f32(a, b3, c3);
    }

    const float* Xrow = x + ((size_t)b * CC) * HWD;
    float*       Yrow = y + ((size_t)b * CC) * HWD;
    const int ncol  = n0 + (lane & 15);
    const int mbase = m0 + 8 * (lane >> 4);
    #pragma unroll
    for (int i = 0; i < 8; ++i) {
        const size_t row = (size_t)(mbase + i) * HWD;
        const float  bi  = bo[mbase + i];
        Yrow[row + ncol +  0] = Xrow[row + ncol +  0] + c0[i] + bi;
        Yrow[row + ncol + 16] = Xrow[row + ncol + 16] + c1[i] + bi;
        Yrow[row + ncol + 32] = Xrow[row + ncol + 32] + c2[i] + bi;
        Yrow[row + ncol + 48] = Xrow[row + ncol + 48] + c3[i] + bi;
    }
}

// ---------------------------------------------------------------------------
extern "C" void kernel_launch(void* const* d_in, const int* in_sizes, int n_in,
                              void* d_out, int out_size, void* d_ws, size_t ws_size,
                              hipStream_t stream)
{
    (void)in_sizes; (void)n_in; (void)out_size; (void)ws_size;

    const float* x  = (const float*)d_in[0];
    const float* wq = (const float*)d_in[1];
    const float* bq = (const float*)d_in[2];
    const float* wk = (const float*)d_in[3];
    const float* bk = (const float*)d_in[4];
    const float* wv = (const float*)d_in[5];
    const float* bv = (const float*)d_in[6];
    const float* wo = (const float*)d_in[7];
    const float* bo = (const float*)d_in[8];

    const size_t tensorElems = (size_t)BB * CC * HWD;   // 4*256*4096 = 4,194,304
    float* q  = (float*)d_ws;
    float* k  = q + tensorElems;
    float* v  = k + tensorElems;
    float* ao = v + tensorElems;

    // 1) fused q/k/v projection: 4 waves/block, each wave 16x32 tile of q,k,v
    dim3 g1(HWD / 128, CC / 16, BB);
    qkv_wmma<<<g1, 128, 0, stream>>>(x, wq, bq, wk, bk, wv, bv, q, k, v);

    // 2) ring attention: one wave per (b, head, pixel) = 65536 waves, 8/block
    ring_attn<<<dim3(BB * NHD * HWD / 8), 256, 0, stream>>>(q, k, v, ao);

    // 3) output projection + residual: each wave 16x64 tile
    dim3 g3(HWD / 256, CC / 16, BB);
    proj_wmma<<<g3, 128, 0, stream>>>(ao, x, wo, bo, (float*)d_out);
}